// TopKGate_927712936391
// MI455X (gfx1250) — compile-verified
//
#include <hip/hip_runtime.h>

#define NTOK 4096
#define MDIM 2048
#define NEXP 32
#define CAP  256
#define SEC  33554432LL   /* NTOK*NEXP*CAP */

typedef __attribute__((ext_vector_type(2))) float v2f;
typedef __attribute__((ext_vector_type(8))) float v8f;

// ---------------------------------------------------------------------------
// Zero the full output (l_aux + combine + dispatch + counts). 256MB of traffic
// -> dominant cost; use b128 stores.
// ---------------------------------------------------------------------------
__global__ void k_zero(float* __restrict__ out, long long n) {
  const long long stride = (long long)gridDim.x * blockDim.x;
  long long i = (long long)blockIdx.x * blockDim.x + threadIdx.x;
  const long long n4 = n >> 2;
  float4* o4 = (float4*)out;
  float4 z; z.x = z.y = z.z = z.w = 0.0f;
  for (long long j = i; j < n4; j += stride) o4[j] = z;
  for (long long j = (n4 << 2) + i; j < n; j += stride) out[j] = 0.0f;
}

__global__ void k_init(float* __restrict__ me_sum, int* __restrict__ counts) {
  const int t = threadIdx.x;
  if (t < NEXP) { me_sum[t] = 0.0f; counts[t] = 0; }
}

// ---------------------------------------------------------------------------
// logits[4096,32] = x[4096,2048] @ wg[32,2048]^T via V_WMMA_F32_16X16X4_F32.
// One wave per 16-row tile; two accumulators cover experts 0-15 / 16-31.
// A 16x4 f32 layout: lanes 0-15 K={0,1}, lanes 16-31 K={2,3} (ISA 7.12.2).
// B 4x16 mirrors A's K split; C/D: VGPR j -> M=j (lanes 0-15) / M=j+8.
// ---------------------------------------------------------------------------
__global__ __launch_bounds__(256) void k_gemm(const float* __restrict__ x,
                                              const float* __restrict__ wg,
                                              float* __restrict__ logits) {
  const int lane  = threadIdx.x & 31;
  const int wave  = (blockIdx.x * blockDim.x + threadIdx.x) >> 5;  // 0..255
  const int m0    = wave * 16;
  const int mrow  = m0 + (lane & 15);
  const int kband = (lane >> 4) * 2;
  const int n     = lane & 15;
  const float* __restrict__ xrow = x  + (size_t)mrow * MDIM;
  const float* __restrict__ w0   = wg + (size_t)n * MDIM;
  const float* __restrict__ w1   = wg + (size_t)(n + 16) * MDIM;

#if __has_builtin(__builtin_amdgcn_wmma_f32_16x16x4_f32)
  v8f c0 = {}; v8f c1 = {};
  #pragma unroll 4
  for (int k = 0; k < MDIM; k += 4) {
    v2f a  = *(const v2f*)(xrow + k + kband);
    v2f b0 = *(const v2f*)(w0 + k + kband);
    v2f b1 = *(const v2f*)(w1 + k + kband);
    c0 = __builtin_amdgcn_wmma_f32_16x16x4_f32(false, a, false, b0, (short)0, c0, false, false);
    c1 = __builtin_amdgcn_wmma_f32_16x16x4_f32(false, a, false, b1, (short)0, c1, false, false);
  }
  const int rbase = m0 + ((lane >> 4) << 3);
  #pragma unroll
  for (int j = 0; j < 8; ++j) {
    logits[(size_t)(rbase + j) * NEXP + n]      = c0[j];
    logits[(size_t)(rbase + j) * NEXP + n + 16] = c1[j];
  }
#else
  // VALU fallback (should not trigger on gfx1250); both wave halves compute
  // identical values, writes are idempotent.
  float acc0 = 0.0f, acc1 = 0.0f;
  for (int k = 0; k < MDIM; ++k) {
    const float xv = xrow[k];
    acc0 = fmaf(xv, w0[k], acc0);
    acc1 = fmaf(xv, w1[k], acc1);
  }
  logits[(size_t)mrow * NEXP + n]      = acc0;
  logits[(size_t)mrow * NEXP + n + 16] = acc1;
#endif
}

// ---------------------------------------------------------------------------
// Per-token gating: one wave32 per token, lane == expert.
// ---------------------------------------------------------------------------
__global__ __launch_bounds__(256) void k_gate(const float* __restrict__ logits,
                                              float* __restrict__ tgcol,
                                              int* __restrict__ top_idx,
                                              float* __restrict__ top_gate,
                                              float* __restrict__ me_sum,
                                              int* __restrict__ counts,
                                              int* __restrict__ keepf) {
  const int lane = threadIdx.x & 31;
  const int tok  = (blockIdx.x * blockDim.x + threadIdx.x) >> 5;
  const float v  = logits[(size_t)tok * NEXP + lane];

  // softmax over 32 experts (wave-wide reductions)
  float m = v;
  for (int off = 16; off; off >>= 1) m = fmaxf(m, __shfl_xor(m, off, 32));
  const float ex = expf(v - m);
  float ssum = ex;
  for (int off = 16; off; off >>= 1) ssum += __shfl_xor(ssum, off, 32);
  const float gate = ex / ssum;

  // top-1 (ties -> smallest index, matching jax.lax.top_k)
  float bv = v; int bi = lane;
  for (int off = 16; off; off >>= 1) {
    const float ov = __shfl_xor(bv, off, 32);
    const int   oi = __shfl_xor(bi, off, 32);
    if (ov > bv || (ov == bv && oi < bi)) { bv = ov; bi = oi; }
  }
  const int i0 = bi;
  // top-2
  float cv = (lane == i0) ? -__builtin_inff() : v;
  int   ci = lane;
  for (int off = 16; off; off >>= 1) {
    const float ov = __shfl_xor(cv, off, 32);
    const int   oi = __shfl_xor(ci, off, 32);
    if (ov > cv || (ov == cv && oi < ci)) { cv = ov; ci = oi; }
  }
  const int i1 = ci;

  const bool chosen = (lane == i0) || (lane == i1);
  // column-major scatter of raw top-k logits (zeros elsewhere) for capacity pass
  tgcol[(size_t)lane * NTOK + tok] = chosen ? v : 0.0f;
  atomicAdd(&me_sum[lane], gate);
  if (chosen) atomicAdd(&counts[lane], 1);

  const float g0 = __shfl(gate, i0, 32);
  const float g1 = __shfl(gate, i1, 32);
  if (lane == 0) {
    top_idx[2 * tok]     = i0;
    top_idx[2 * tok + 1] = i1;
    top_gate[2 * tok]     = g0;
    top_gate[2 * tok + 1] = g1;
  }
  if (lane < 2) keepf[2 * tok + lane] = 0;
}

// block-wide exclusive scan over 256 values (Hillis-Steele in LDS)
__device__ __forceinline__ unsigned excl_scan_256(unsigned local, unsigned* sbuf, int tid) {
  sbuf[tid] = local;
  __syncthreads();
  for (int off = 1; off < 256; off <<= 1) {
    const unsigned v = (tid >= off) ? sbuf[tid - off] : 0u;
    __syncthreads();
    sbuf[tid] += v;
    __syncthreads();
  }
  const unsigned incl = sbuf[tid];
  __syncthreads();
  return incl - local;
}

// ---------------------------------------------------------------------------
// Per-expert capacity: exact top-CAP-of-NTOK via 4-pass radix select on
// order-preserving uint keys, stable tie-break by token index; then exclusive
// scan of the final mask to produce capacity slot locations.
// ---------------------------------------------------------------------------
__global__ __launch_bounds__(256) void k_expert_cap(const float* __restrict__ tgcol,
                                                    const int* __restrict__ top_idx,
                                                    int* __restrict__ keepf,
                                                    int* __restrict__ locarr) {
  __shared__ float    col[NTOK];
  __shared__ unsigned hist[256];
  __shared__ unsigned sbuf[256];
  __shared__ unsigned sh_sel, sh_rem;
  const int e   = blockIdx.x;
  const int tid = threadIdx.x;

  for (int i = tid; i < NTOK; i += 256) col[i] = tgcol[(size_t)e * NTOK + i];
  __syncthreads();

  unsigned key[16];
  #pragma unroll
  for (int i = 0; i < 16; ++i) {
    const unsigned u = __float_as_uint(col[tid * 16 + i]);
    key[i] = (u & 0x80000000u) ? ~u : (u | 0x80000000u);  // ascending == float order
  }

  unsigned prefix = 0, highmask = 0, krem = CAP;
  for (int byte = 3; byte >= 0; --byte) {
    const int shift = byte * 8;
    hist[tid] = 0;
    __syncthreads();
    #pragma unroll
    for (int i = 0; i < 16; ++i)
      if ((key[i] & highmask) == prefix) atomicAdd(&hist[(key[i] >> shift) & 0xFFu], 1u);
    __syncthreads();
    if (tid == 0) {
      unsigned cum = 0; sh_sel = 0; sh_rem = krem;
      for (int bb = 255; bb >= 0; --bb) {
        const unsigned c = hist[bb];
        if (cum + c >= krem) { sh_sel = (unsigned)bb; sh_rem = krem - cum; break; }
        cum += c;
      }
    }
    __syncthreads();
    prefix   |= sh_sel << shift;
    highmask |= 0xFFu << shift;
    krem      = sh_rem;
    __syncthreads();
  }
  const unsigned T        = prefix;  // key of the CAP-th largest element
  const unsigned keepTies = krem;    // #elements equal to T to keep (lowest idx first)

  // stable rank among ties (token order)
  unsigned tflag[16]; unsigned tsum = 0;
  #pragma unroll
  for (int i = 0; i < 16; ++i) { tflag[i] = (key[i] == T) ? 1u : 0u; tsum += tflag[i]; }
  unsigned trun = excl_scan_256(tsum, sbuf, tid);

  // final mask = capacity-keep AND routed (nonzero entry)
  unsigned kflag[16]; unsigned ksum = 0;
  #pragma unroll
  for (int i = 0; i < 16; ++i) {
    const bool capkeep = (key[i] > T) || (tflag[i] && trun < keepTies);
    trun += tflag[i];
    kflag[i] = (capkeep && (col[tid * 16 + i] != 0.0f)) ? 1u : 0u;
    ksum += kflag[i];
  }

  // locations = cumsum(mask) - 1  (exclusive scan over token order)
  unsigned run = excl_scan_256(ksum, sbuf, tid);
  #pragma unroll
  for (int i = 0; i < 16; ++i) {
    if (kflag[i]) {
      const int s    = tid * 16 + i;
      const int slot = (top_idx[2 * s] == e) ? 0 : 1;
      keepf[2 * s + slot]  = 1;
      locarr[2 * s + slot] = (int)run;
    }
    run += kflag[i];
  }
}

// ---------------------------------------------------------------------------
// Per-token finalize: renormalize kept gates, shared slot = sum of kept
// locations (reference's locations_s), scatter combine/dispatch; l_aux+counts.
// ---------------------------------------------------------------------------
__global__ __launch_bounds__(256) void k_finalize(const int* __restrict__ top_idx,
                                                  const float* __restrict__ top_gate,
                                                  const int* __restrict__ keepf,
                                                  const int* __restrict__ locarr,
                                                  const float* __restrict__ me_sum,
                                                  const int* __restrict__ counts,
                                                  float* __restrict__ out) {
  const int s = blockIdx.x * blockDim.x + threadIdx.x;
  if (s < NTOK) {
    const int   i0 = top_idx[2 * s],  i1 = top_idx[2 * s + 1];
    const float g0 = top_gate[2 * s], g1 = top_gate[2 * s + 1];
    const int   f0 = keepf[2 * s],    f1 = keepf[2 * s + 1];
    float denom = (f0 ? g0 : 0.0f) + (f1 ? g1 : 0.0f);
    denom = fmaxf(denom, 1.1920929e-07f);  // f32 eps clip
    const int csum = (f0 ? locarr[2 * s] : 0) + (f1 ? locarr[2 * s + 1] : 0);
    if (csum < CAP) {  // one_hot drops out-of-range slots
      const long long base = 1LL + (long long)s * (NEXP * CAP);
      if (f0) {
        const long long off = base + (long long)i0 * CAP + csum;
        out[off]       = g0 / denom;   // combine_weights
        out[off + SEC] = 1.0f;         // dispatch_mask
      }
      if (f1) {
        const long long off = base + (long long)i1 * CAP + csum;
        out[off]       = g1 / denom;
        out[off + SEC] = 1.0f;
      }
    }
  }
  if (blockIdx.x == 0) {
    if (threadIdx.x == 0) {
      float acc = 0.0f;
      for (int e = 0; e < NEXP; ++e)
        acc += (me_sum[e] / (float)NTOK) * ((float)counts[e] / (float)NTOK);
      out[0] = acc * ((float)NEXP / 2.0f);  // mean_e(me*ce)*E*E/K == sum*E/K
    }
    if (threadIdx.x < NEXP)
      out[1 + 2 * SEC + threadIdx.x] = (float)counts[threadIdx.x];
  }
}

extern "C" void kernel_launch(void* const* d_in, const int* in_sizes, int n_in,
                              void* d_out, int out_size, void* d_ws, size_t ws_size,
                              hipStream_t stream) {
  const float* x  = (const float*)d_in[0];
  const float* wg = (const float*)d_in[1];
  float* out = (float*)d_out;
  char*  ws  = (char*)d_ws;

  float* logits   = (float*)(ws);              // 4096*32 f32      = 512 KB
  float* tgcol    = (float*)(ws + 524288);     // 32*4096 f32      = 512 KB
  int*   top_idx  = (int*)  (ws + 1048576);    // 4096*2 i32
  float* top_gate = (float*)(ws + 1081344);    // 4096*2 f32
  float* me_sum   = (float*)(ws + 1114112);    // 32 f32
  int*   counts   = (int*)  (ws + 1114240);    // 32 i32
  int*   keepf    = (int*)  (ws + 1114368);    // 4096*2 i32
  int*   locarr   = (int*)  (ws + 1147136);    // 4096*2 i32

  const long long total = 1LL + 2LL * SEC + NEXP;
  k_zero<<<4096, 256, 0, stream>>>(out, total);
  k_init<<<1, 64, 0, stream>>>(me_sum, counts);
  k_gemm<<<32, 256, 0, stream>>>(x, wg, logits);
  k_gate<<<512, 256, 0, stream>>>(logits, tgcol, top_idx, top_gate, me_sum, counts, keepf);
  k_expert_cap<<<NEXP, 256, 0, stream>>>(tgcol, top_idx, keepf, locarr);
  k_finalize<<<NTOK / 256, 256, 0, stream>>>(top_idx, top_gate, keepf, locarr, me_sum, counts, out);
}